// GraphConvolution_56556129354712
// MI455X (gfx1250) — compile-verified
//
#include <hip/hip_runtime.h>

typedef __attribute__((ext_vector_type(2))) float v2f;
typedef __attribute__((ext_vector_type(8))) float v8f;

#define F_IN  128
#define F_OUT 64
// W repacked in LDS: K-pair p (=k>>1) at word p*PAIR_STRIDE, column c at +c*2,
// parity (k&1) at +0/+1.  PAIR_STRIDE=160 => half-wave K-pair delta = 160 words
// = +32 banks mod 64, so a wave-wide ds_load_b64 (lanes 0-15 pair p, lanes
// 16-31 pair p+1) covers all 64 banks conflict-free.
#define PAIR_STRIDE 160

// ---------------------------------------------------------------------------
// support[N,64] = X[N,128] @ W[128,64] via V_WMMA_F32_16X16X4_F32.
// One wave per 16-row strip; 4 accumulators cover all 64 output columns, so X
// is streamed exactly once (memory-bound optimum). Each WMMA operand (A and
// all four Bs) is a single contiguous b64 load -> no repacking movs.
// ---------------------------------------------------------------------------
__global__ __launch_bounds__(256) void gcn_gemm_wmma(const float* __restrict__ x,
                                                     const float* __restrict__ w,
                                                     float* __restrict__ support,
                                                     int n_nodes) {
    __shared__ float lds_w[(F_IN / 2) * PAIR_STRIDE]; // 40 KB of 320 KB WGP LDS
    const int tid = threadIdx.x;
    for (int i = tid; i < F_IN * F_OUT; i += 256) {
        const int k = i >> 6, c = i & 63;
        lds_w[(k >> 1) * PAIR_STRIDE + c * 2 + (k & 1)] = w[i];
    }
    __syncthreads();

    const int wave   = tid >> 5;
    const int lane   = tid & 31;
    const int lane15 = lane & 15;
    const int koff   = (lane >> 4) << 1;   // lanes 0-15 hold K={0,1}, 16-31 hold K={2,3}

    const int strip = blockIdx.x * 8 + wave;
    if (strip >= (n_nodes >> 4)) return;   // wave-uniform: EXEC stays all-1s for WMMA
    const long row_base = (long)strip * 16;

    const float* xrow = x + (row_base + lane15) * F_IN;

    v8f acc0 = {}, acc1 = {}, acc2 = {}, acc3 = {};
    for (int k = 0; k < F_IN; k += 4) {
        // A operand: X[row][k+koff .. k+koff+1], contiguous -> global_load_b64
        const v2f a = *(const v2f*)(xrow + k + koff);

        // B operands: W rows (k+koff, k+koff+1) pair-interleaved -> ds_load_b64
        const int base = ((k + koff) >> 1) * PAIR_STRIDE + lane15 * 2;
        const v2f b0 = *(const v2f*)&lds_w[base +  0];  // cols  0..15
        const v2f b1 = *(const v2f*)&lds_w[base + 32];  // cols 16..31
        const v2f b2 = *(const v2f*)&lds_w[base + 64];  // cols 32..47
        const v2f b3 = *(const v2f*)&lds_w[base + 96];  // cols 48..63

        acc0 = __builtin_amdgcn_wmma_f32_16x16x4_f32(false, a, false, b0, (short)0, acc0, false, false);
        acc1 = __builtin_amdgcn_wmma_f32_16x16x4_f32(false, a, false, b1, (short)0, acc1, false, false);
        acc2 = __builtin_amdgcn_wmma_f32_16x16x4_f32(false, a, false, b2, (short)0, acc2, false, false);
        acc3 = __builtin_amdgcn_wmma_f32_16x16x4_f32(false, a, false, b3, (short)0, acc3, false, false);
    }

    // C/D layout: VGPR r holds M=r (lanes 0-15) and M=r+8 (lanes 16-31), N=lane15
    const int rowadd = (lane >> 4) << 3;
    float* outp = support + (row_base + rowadd) * F_OUT + lane15;
#pragma unroll
    for (int r = 0; r < 8; ++r) {
        outp[(long)r * F_OUT + 0]  = acc0[r];
        outp[(long)r * F_OUT + 16] = acc1[r];
        outp[(long)r * F_OUT + 32] = acc2[r];
        outp[(long)r * F_OUT + 48] = acc3[r];
    }
}

// out[i][f] = bias[f]  (segment_sum accumulates on top of this)
__global__ void gcn_bias_init(float* __restrict__ out, const float* __restrict__ bias,
                              int total) {
    int idx = blockIdx.x * blockDim.x + threadIdx.x;
    if (idx < total) out[idx] = bias[idx & (F_OUT - 1)];
}

// ---------------------------------------------------------------------------
// COO scatter: one wave per edge. Edge metadata loads are wave-broadcast (one
// request); each lane handles a float2 of the 64-feature row (coalesced 256B
// gather, L2-resident since support is 25.6MB < 192MB L2) and issues two
// hardware f32 atomic adds (no-return / STOREcnt path, resolved in L2).
// ---------------------------------------------------------------------------
__global__ __launch_bounds__(256) void gcn_edge_scatter(const float* __restrict__ support,
                                                        const int*   __restrict__ esrc,
                                                        const int*   __restrict__ edst,
                                                        const float* __restrict__ ew,
                                                        float* __restrict__ out,
                                                        int n_edges) {
    const int e    = (blockIdx.x * 256 + threadIdx.x) >> 5;
    const int lane = threadIdx.x & 31;
    if (e >= n_edges) return;

    const int   src = esrc[e];
    const int   dst = edst[e];
    const float wgt = ew[e];

    const float2 v = *(const float2*)(support + (long)src * F_OUT + lane * 2);
    float* o = out + (long)dst * F_OUT + lane * 2;
    unsafeAtomicAdd(o + 0, v.x * wgt);
    unsafeAtomicAdd(o + 1, v.y * wgt);
}

extern "C" void kernel_launch(void* const* d_in, const int* in_sizes, int n_in,
                              void* d_out, int out_size, void* d_ws, size_t ws_size,
                              hipStream_t stream) {
    const float* x           = (const float*)d_in[0];
    const float* edge_weight = (const float*)d_in[1];
    const float* weight      = (const float*)d_in[2];
    const float* bias        = (const float*)d_in[3];
    const int*   edge_src    = (const int*)d_in[4];
    const int*   edge_dst    = (const int*)d_in[5];
    float* out     = (float*)d_out;
    float* support = (float*)d_ws;            // 100000*64*4 = 25.6 MB scratch

    const int n_nodes = in_sizes[0] / F_IN;
    const int n_edges = in_sizes[1];

    // 1) support = X @ W  (8 waves/block, 1 strip of 16 rows per wave)
    const int strips      = n_nodes >> 4;
    const int gemm_blocks = (strips + 7) / 8;
    gcn_gemm_wmma<<<gemm_blocks, 256, 0, stream>>>(x, weight, support, n_nodes);

    // 2) out = broadcast(bias)
    const int total = n_nodes * F_OUT;
    gcn_bias_init<<<(total + 255) / 256, 256, 0, stream>>>(out, bias, total);

    // 3) out[dst] += support[src] * w_e   (wave per edge)
    const int sc_blocks = (n_edges + 7) / 8;
    gcn_edge_scatter<<<sc_blocks, 256, 0, stream>>>(support, edge_src, edge_dst,
                                                    edge_weight, out, n_edges);
}